// PermutationInvariantLinear_58351425683886
// MI455X (gfx1250) — compile-verified
//
#include <hip/hip_runtime.h>
#include <stdint.h>

#define IN_F 64
#define OUT_F 64
#define CODE_ZERO 666
#define LDS_STRIDE 68               // 64 + 4 pad floats -> conflict-free frag reads
#define ROWS_PER_BLOCK 128          // 8 waves x 16 rows
#define THREADS 256

typedef __attribute__((ext_vector_type(2))) float v2f;
typedef __attribute__((ext_vector_type(4))) float v4f;
typedef __attribute__((ext_vector_type(8))) float v8f;

// ---------------------------------------------------------------------------
// gfx1250 async copy: global -> LDS (16 B per lane), tracked by ASYNCcnt.
// Low 32 bits of a generic shared-memory pointer are the LDS byte offset
// (flat aperture places LDS offset in addr[31:0]).
// ---------------------------------------------------------------------------
__device__ __forceinline__ void async_copy_b128(const float* gsrc, float* ldst) {
    uint32_t lds_addr = (uint32_t)(uintptr_t)ldst;
    uint64_t gaddr    = (uint64_t)(uintptr_t)gsrc;
    asm volatile("global_load_async_to_lds_b128 %0, %1, off"
                 :
                 : "v"(lds_addr), "v"(gaddr)
                 : "memory");
}

__device__ __forceinline__ void wait_async_all() {
#if __has_builtin(__builtin_amdgcn_s_wait_asynccnt)
    __builtin_amdgcn_s_wait_asynccnt(0);
#else
    asm volatile("s_wait_asynccnt 0x0" ::: "memory");
#endif
}

// ---------------------------------------------------------------------------
// Step 1: W_eff[o,i] = (perm[i]==666) ? 0 : signs[i]*W[o, clamp(perm[i],0,63)]
// ---------------------------------------------------------------------------
__global__ void build_weff_kernel(const float* __restrict__ w,
                                  const float* __restrict__ signs,
                                  const int* __restrict__ perm,
                                  float* __restrict__ weff) {
    int i = threadIdx.x;   // input column 0..63
    int o = blockIdx.x;    // output row 0..63
    int p = perm[i];
    int sp = p < 0 ? 0 : (p > (IN_F - 1) ? (IN_F - 1) : p);
    float v = (p == CODE_ZERO) ? 0.0f : signs[i] * w[o * IN_F + sp];
    weff[o * IN_F + i] = v;
}

// ---------------------------------------------------------------------------
// Step 2: out = x @ W_eff^T + bias via V_WMMA_F32_16X16X4_F32
// ---------------------------------------------------------------------------
__global__ __launch_bounds__(THREADS) void pil_wmma_kernel(
    const float* __restrict__ x,
    const float* __restrict__ weff,
    const float* __restrict__ bias,
    float* __restrict__ out) {

    __shared__ float lds_w[OUT_F * LDS_STRIDE];           // 17,408 B
    __shared__ float lds_x[ROWS_PER_BLOCK * LDS_STRIDE];  // 34,816 B

    const int tid = threadIdx.x;

    // Stage W_eff: 64 rows x 16 float4 = 1024 vec4 copies, 4 per thread,
    // coalesced global -> LDS async DMA (no VGPR round trip).
    #pragma unroll
    for (int r = 0; r < 4; ++r) {
        int idx = tid + r * THREADS;       // 0..1023
        int row = idx >> 4;                // 0..63
        int c4  = (idx & 15) << 2;         // 0,4,...,60
        async_copy_b128(weff + row * IN_F + c4, lds_w + row * LDS_STRIDE + c4);
    }

    // Stage x tile: 128 rows x 16 float4 = 2048 vec4 copies, 8 per thread.
    const long long row0 = (long long)blockIdx.x * ROWS_PER_BLOCK;
    const float* xblk = x + row0 * IN_F;
    #pragma unroll
    for (int r = 0; r < 8; ++r) {
        int idx = tid + r * THREADS;       // 0..2047
        int row = idx >> 4;                // 0..127
        int c4  = (idx & 15) << 2;
        async_copy_b128(xblk + (long long)row * IN_F + c4,
                        lds_x + row * LDS_STRIDE + c4);
    }

    wait_async_all();      // this wave's async copies are in LDS
    __syncthreads();       // all waves' copies visible

    const int wave  = tid >> 5;            // 0..7 -> rows [wave*16, wave*16+16)
    const int lane  = tid & 31;
    const int lrow  = lane & 15;           // M (A) / N (B,D) sub-index
    const int khalf = (lane >> 4) << 1;    // 0 or 2: K offset for upper half-wave

    // Accumulators: 4 N-tiles of 16x16, C initialized with bias[n].
    v8f acc[4];
    #pragma unroll
    for (int nt = 0; nt < 4; ++nt) {
        float b = bias[nt * 16 + lrow];
        acc[nt] = (v8f){b, b, b, b, b, b, b, b};
    }

    const float* aptr  = lds_x + (wave * 16 + lrow) * LDS_STRIDE + khalf;
    const float* bbase = lds_w + lrow * LDS_STRIDE + khalf;

    #pragma unroll
    for (int k0 = 0; k0 < IN_F; k0 += 4) {
        // A fragment (16x4 f32): lane holds x[M, k0+khalf .. +1] -> aligned float2
        v2f a = *(const v2f*)(aptr + k0);
        #pragma unroll
        for (int nt = 0; nt < 4; ++nt) {
            // B fragment (4x16 f32): lane holds W_eff[N, k0+khalf .. +1]
            v2f b = *(const v2f*)(bbase + nt * 16 * LDS_STRIDE + k0);
            acc[nt] = __builtin_amdgcn_wmma_f32_16x16x4_f32(
                /*neg_a=*/false, a, /*neg_b=*/false, b,
                /*c_mod=*/(short)0, acc[nt],
                /*reuse_a=*/false, /*reuse_b=*/false);
        }
    }

    // Store D: VGPR v -> M = v + (lane/16)*8, N = lrow.
    float* oblk = out + (row0 + wave * 16) * OUT_F;
    const int moff = (lane >> 4) << 3;     // 0 or 8
    #pragma unroll
    for (int nt = 0; nt < 4; ++nt) {
        #pragma unroll
        for (int v = 0; v < 8; ++v) {
            oblk[(long long)(v + moff) * OUT_F + nt * 16 + lrow] = acc[nt][v];
        }
    }
}

// ---------------------------------------------------------------------------
// Launch
// ---------------------------------------------------------------------------
extern "C" void kernel_launch(void* const* d_in, const int* in_sizes, int n_in,
                              void* d_out, int out_size, void* d_ws, size_t ws_size,
                              hipStream_t stream) {
    const float* x      = (const float*)d_in[0];
    const float* weight = (const float*)d_in[1];
    const float* bias   = (const float*)d_in[2];
    const float* signs  = (const float*)d_in[3];
    const int*   perm   = (const int*)d_in[4];

    float* weff = (float*)d_ws;            // 64*64*4 = 16 KB scratch
    float* out  = (float*)d_out;

    const long long batch = (long long)in_sizes[0] / IN_F;

    build_weff_kernel<<<dim3(OUT_F), dim3(IN_F), 0, stream>>>(weight, signs, perm, weff);

    const int nblocks = (int)(batch / ROWS_PER_BLOCK);
    pil_wmma_kernel<<<dim3(nblocks), dim3(THREADS), 0, stream>>>(x, weff, bias, out);
}